// GCN_3298534884297
// MI455X (gfx1250) — compile-verified
//
#include <hip/hip_runtime.h>

#define N_NODES 100000
#define N_EDGES 800000
#define D_IN    16
#define H       128
#define OUT_DIM 3
#define NEG_SLOPE 0.01f

#define LDW   (H + 2)      // padded col stride for 128-deep W  (bank-conflict-free b64)
#define LDWIN (D_IN + 2)   // padded col stride for 16-deep W_in

typedef float v2f __attribute__((ext_vector_type(2)));
typedef float v8f __attribute__((ext_vector_type(8)));

static __device__ __forceinline__ v8f wmma_f32_k4(v2f a, v2f b, v8f c) {
    // D = A(16x4,f32) * B(4x16,f32) + C(16x16,f32)
    return __builtin_amdgcn_wmma_f32_16x16x4_f32(false, a, false, b, (short)0, c,
                                                 false, false);
}

// ---------------------------------------------------------------------------
// x0 = leaky_relu(feature @ W_in + b_in)   [N,16] x [16,128]
// W_in staged TRANSPOSED in LDS: Wt[col*LDWIN + k] -> B fragment is one b64
// ---------------------------------------------------------------------------
__global__ __launch_bounds__(128) void k_lin_in(const float* __restrict__ feat,
                                                const float* __restrict__ Win,
                                                const float* __restrict__ bin,
                                                float* __restrict__ out) {
    __shared__ float Wt[H * LDWIN];   // 9 KB, transposed + padded
    __shared__ float bl[H];
    const int tid = threadIdx.x;
    for (int idx = tid; idx < D_IN * H / 4; idx += 128) {
        const int k  = idx >> 5;          // row of W_in (K index)
        const int c4 = idx & 31;          // float4 column block
        const float4 v = ((const float4*)Win)[idx];
        Wt[(c4 * 4 + 0) * LDWIN + k] = v.x;
        Wt[(c4 * 4 + 1) * LDWIN + k] = v.y;
        Wt[(c4 * 4 + 2) * LDWIN + k] = v.z;
        Wt[(c4 * 4 + 3) * LDWIN + k] = v.w;
    }
    if (tid < H) bl[tid] = bin[tid];
    __syncthreads();

    const int wave = tid >> 5, lane = tid & 31;
    const int half = lane >> 4, l16 = lane & 15;
    const int nTiles = N_NODES / 16;
    int tile = blockIdx.x * 4 + wave;
    const bool active = tile < nTiles;
    if (!active) tile = nTiles - 1;          // WMMA needs full EXEC: clamp, skip store

    const float* fr = feat + (size_t)(tile * 16 + l16) * D_IN;

    v8f acc[8];
    const v8f z = {0.f,0.f,0.f,0.f,0.f,0.f,0.f,0.f};
#pragma unroll
    for (int nt = 0; nt < 8; ++nt) acc[nt] = z;

#pragma unroll
    for (int k0 = 0; k0 < D_IN; k0 += 4) {
        const int kb = k0 + 2 * half;
        v2f a = *(const v2f*)(fr + kb);
#pragma unroll
        for (int nt = 0; nt < 8; ++nt) {
            const int col = nt * 16 + l16;
            v2f b = *(const v2f*)(&Wt[col * LDWIN + kb]);   // single ds_load_b64
            acc[nt] = wmma_f32_k4(a, b, acc[nt]);
        }
    }
    if (active) {
#pragma unroll
        for (int nt = 0; nt < 8; ++nt) {
            const int col = nt * 16 + l16;
            const float bias = bl[col];
#pragma unroll
            for (int r = 0; r < 8; ++r) {
                const int m = tile * 16 + half * 8 + r;
                float v = acc[nt][r] + bias;
                out[(size_t)m * H + col] = v > 0.f ? v : NEG_SLOPE * v;
            }
        }
    }
}

// ---------------------------------------------------------------------------
// h = x @ W    [N,128] x [128,128]
// W staged TRANSPOSED+padded in LDS (66.5 KB of the WGP's 320 KB):
// B fragment {W[kb][col], W[kb+1][col]} = contiguous b64, banks 2c..2c+1
// ---------------------------------------------------------------------------
__global__ __launch_bounds__(128) void k_gemm128(const float* __restrict__ x,
                                                 const float* __restrict__ W,
                                                 float* __restrict__ out) {
    __shared__ float Wt[H * LDW];
    const int tid = threadIdx.x;
    for (int idx = tid; idx < H * H / 4; idx += 128) {
        const int k  = idx >> 5;          // row of W (K index), 32 float4 per row
        const int c4 = idx & 31;
        const float4 v = ((const float4*)W)[idx];
        Wt[(c4 * 4 + 0) * LDW + k] = v.x;
        Wt[(c4 * 4 + 1) * LDW + k] = v.y;
        Wt[(c4 * 4 + 2) * LDW + k] = v.z;
        Wt[(c4 * 4 + 3) * LDW + k] = v.w;
    }
    __syncthreads();

    const int wave = tid >> 5, lane = tid & 31;
    const int half = lane >> 4, l16 = lane & 15;
    const int nTiles = N_NODES / 16;
    int tile = blockIdx.x * 4 + wave;
    const bool active = tile < nTiles;
    if (!active) tile = nTiles - 1;

    const float* xr = x + (size_t)(tile * 16 + l16) * H;

    v8f acc[8];
    const v8f z = {0.f,0.f,0.f,0.f,0.f,0.f,0.f,0.f};
#pragma unroll
    for (int nt = 0; nt < 8; ++nt) acc[nt] = z;

    for (int k0 = 0; k0 < H; k0 += 4) {
        const int kb = k0 + 2 * half;
        v2f a = *(const v2f*)(xr + kb);
#pragma unroll
        for (int nt = 0; nt < 8; ++nt) {
            const int col = nt * 16 + l16;
            v2f b = *(const v2f*)(&Wt[col * LDW + kb]);     // single ds_load_b64
            acc[nt] = wmma_f32_k4(a, b, acc[nt]);
        }
    }
    if (active) {
#pragma unroll
        for (int nt = 0; nt < 8; ++nt) {
            const int col = nt * 16 + l16;
#pragma unroll
            for (int r = 0; r < 8; ++r) {
                const int m = tile * 16 + half * 8 + r;
                out[(size_t)m * H + col] = acc[nt][r];
            }
        }
    }
}

// ---------------------------------------------------------------------------
// degree (with self loop), then dinv = rsqrt(deg) in place
// ---------------------------------------------------------------------------
__global__ void k_deg_init(float* __restrict__ deg) {
    int i = blockIdx.x * blockDim.x + threadIdx.x;
    if (i < N_NODES) deg[i] = 1.0f;          // self loop
}
__global__ void k_deg_edge(const int* __restrict__ ei, float* __restrict__ deg) {
    int e = blockIdx.x * blockDim.x + threadIdx.x;
    if (e < N_EDGES) atomicAdd(&deg[ei[N_EDGES + e]], 1.0f);   // dst degree
}
__global__ void k_rsqrt(float* __restrict__ deg) {
    int i = blockIdx.x * blockDim.x + threadIdx.x;
    if (i < N_NODES) deg[i] = rsqrtf(fmaxf(deg[i], 1.0f));
}

// ---------------------------------------------------------------------------
// out[i,:] = b + h[i,:] * dinv[i]^2   (self-loop message + bias)
// ---------------------------------------------------------------------------
__global__ void k_agg_init(const float* __restrict__ h, const float* __restrict__ dinv,
                           const float* __restrict__ b, float* __restrict__ out) {
    int idx = blockIdx.x * blockDim.x + threadIdx.x;   // one float4 each
    if (idx >= N_NODES * (H / 4)) return;
    const int node = idx >> 5;                          // H/4 == 32
    const int j4   = idx & 31;
    float w = dinv[node]; w *= w;
    const float4 hv = ((const float4*)h)[idx];
    const float4 bv = ((const float4*)b)[j4];
    float4 o;
    o.x = bv.x + hv.x * w;
    o.y = bv.y + hv.y * w;
    o.z = bv.z + hv.z * w;
    o.w = bv.w + hv.w * w;
    ((float4*)out)[idx] = o;
}

// ---------------------------------------------------------------------------
// edges: out[dst,:] += h[src,:] * dinv[src]*dinv[dst]   (32 threads/edge)
// ---------------------------------------------------------------------------
__global__ void k_edge(const float* __restrict__ h, const float* __restrict__ dinv,
                       const int* __restrict__ ei, float* __restrict__ out) {
    int gid = blockIdx.x * blockDim.x + threadIdx.x;
    if (gid >= N_EDGES * 32) return;
    const int e  = gid >> 5;
    const int j4 = gid & 31;
    const int s = ei[e];
    const int d = ei[N_EDGES + e];
    const float w = dinv[s] * dinv[d];
    const float4 hv = ((const float4*)(h + (size_t)s * H))[j4];
    float* op = out + (size_t)d * H + j4 * 4;
    atomicAdd(op + 0, hv.x * w);
    atomicAdd(op + 1, hv.y * w);
    atomicAdd(op + 2, hv.z * w);
    atomicAdd(op + 3, hv.w * w);
}

// ---------------------------------------------------------------------------
// y = x @ W_out + b_out   [N,128] x [128,3]  (tiny N-dim: plain VALU)
// ---------------------------------------------------------------------------
__global__ void k_lin_out(const float* __restrict__ x, const float* __restrict__ Wout,
                          const float* __restrict__ bout, float* __restrict__ out) {
    int i = blockIdx.x * blockDim.x + threadIdx.x;
    if (i >= N_NODES) return;
    float a0 = bout[0], a1 = bout[1], a2 = bout[2];
    const float* xr = x + (size_t)i * H;
    for (int k = 0; k < H; k += 4) {
        const float4 xv = *(const float4*)(xr + k);
        a0 += xv.x * Wout[(k + 0) * 3 + 0] + xv.y * Wout[(k + 1) * 3 + 0]
            + xv.z * Wout[(k + 2) * 3 + 0] + xv.w * Wout[(k + 3) * 3 + 0];
        a1 += xv.x * Wout[(k + 0) * 3 + 1] + xv.y * Wout[(k + 1) * 3 + 1]
            + xv.z * Wout[(k + 2) * 3 + 1] + xv.w * Wout[(k + 3) * 3 + 1];
        a2 += xv.x * Wout[(k + 0) * 3 + 2] + xv.y * Wout[(k + 1) * 3 + 2]
            + xv.z * Wout[(k + 2) * 3 + 2] + xv.w * Wout[(k + 3) * 3 + 2];
    }
    out[(size_t)i * 3 + 0] = a0;
    out[(size_t)i * 3 + 1] = a1;
    out[(size_t)i * 3 + 2] = a2;
}

extern "C" void kernel_launch(void* const* d_in, const int* in_sizes, int n_in,
                              void* d_out, int out_size, void* d_ws, size_t ws_size,
                              hipStream_t stream) {
    (void)in_sizes; (void)n_in; (void)out_size; (void)ws_size;
    const float* feat = (const float*)d_in[0];
    const int*   ei   = (const int*)d_in[1];   // [2,E]: row0=src, row1=dst
    /* d_in[2] = edge_type: unused, as in the reference */
    const float* Win  = (const float*)d_in[3];
    const float* bin  = (const float*)d_in[4];
    const float* W1   = (const float*)d_in[5];
    const float* b1   = (const float*)d_in[6];
    const float* W2   = (const float*)d_in[7];
    const float* b2   = (const float*)d_in[8];
    const float* Wout = (const float*)d_in[9];
    const float* bout = (const float*)d_in[10];
    float* out = (float*)d_out;

    float* ws = (float*)d_ws;
    const size_t NH = (size_t)N_NODES * H;
    float* buf0 = ws;                 // x0 / h2
    float* buf1 = ws + NH;            // h1 / y2
    float* buf2 = ws + 2 * NH;        // y1 (= x for conv2)
    float* deg  = ws + 3 * NH;        // deg -> dinv in place

    const int nTiles = N_NODES / 16;                 // 6250
    const dim3 gT((nTiles + 3) / 4), bT(128);        // 4 waves/block, 1 tile/wave
    const int bN  = (N_NODES + 255) / 256;
    const int bE  = (N_EDGES + 255) / 256;
    const int bNH = (N_NODES * 32 + 255) / 256;      // float4 granularity
    const int bEH = (N_EDGES * 32 + 255) / 256;

    // x0 = leaky_relu(feature @ W_in + b_in)
    k_lin_in<<<gT, bT, 0, stream>>>(feat, Win, bin, buf0);

    // symmetric normalization coefficients
    k_deg_init<<<bN, 256, 0, stream>>>(deg);
    k_deg_edge<<<bE, 256, 0, stream>>>(ei, deg);
    k_rsqrt  <<<bN, 256, 0, stream>>>(deg);

    // conv1: h1 = x0@W1 ; y1 = scatter(h1) + b1
    k_gemm128 <<<gT, bT, 0, stream>>>(buf0, W1, buf1);
    k_agg_init<<<bNH, 256, 0, stream>>>(buf1, deg, b1, buf2);
    k_edge    <<<bEH, 256, 0, stream>>>(buf1, deg, ei, buf2);

    // conv2: h2 = y1@W2 ; y2 = scatter(h2) + b2
    k_gemm128 <<<gT, bT, 0, stream>>>(buf2, W2, buf0);
    k_agg_init<<<bNH, 256, 0, stream>>>(buf0, deg, b2, buf1);
    k_edge    <<<bEH, 256, 0, stream>>>(buf0, deg, ei, buf1);

    // final projection
    k_lin_out<<<bN, 256, 0, stream>>>(buf1, Wout, bout, out);
}